// LengthRegulator_70042326664005
// MI455X (gfx1250) — compile-verified
//
#include <hip/hip_runtime.h>

// Problem constants from the reference: B=32, T=1024, C=1024, MAX_LEN=2048.
#define B_      32
#define T_      1024
#define C_      1024
#define C4_     (C_ / 4)       // 256 float4 per row
#define MAXLEN_ 2048
#define ROWS_PER_BLOCK 4

// b128 payload type the async builtins expect (per hipcc diagnostics):
//   arg0: generic  "__vector_size__(16) int *"           (global side)
//   arg1: shared   "__vector_size__(16) int __shared__*" (LDS side)
typedef int b128_t __attribute__((vector_size(16)));
typedef b128_t __attribute__((address_space(3))) * lds_b128_ptr;

// Native clang vector for __builtin_nontemporal_store (HIP float4 is a struct
// and is rejected by the builtin).
typedef float f32x4 __attribute__((ext_vector_type(4)));

// Store temporal hint NT (ISA store TH table: 1 = non-temporal). Output is
// write-once/streaming; NT keeps L2 free so x (128MB) stays resident in the
// 192MB L2 and duration-2 re-reads hit L2 instead of HBM.
#define CPOL_STORE_NT 1

#if defined(__gfx1250__) && __has_builtin(__builtin_amdgcn_global_load_async_to_lds_b128) && \
    __has_builtin(__builtin_amdgcn_global_store_async_from_lds_b128)
#define HAS_ASYNC_LDS 1
#else
#define HAS_ASYNC_LDS 0
#endif

#if defined(__gfx1250__) && __has_builtin(__builtin_amdgcn_s_wait_asynccnt)
#define WAIT_ASYNCCNT0() __builtin_amdgcn_s_wait_asynccnt(0)
#elif defined(__gfx1250__)
#define WAIT_ASYNCCNT0() asm volatile("s_wait_asynccnt 0" ::: "memory")
#else
#define WAIT_ASYNCCNT0()
#endif

// ---------------- Kernel 1: masked inclusive scan of durations ----------------
__global__ __launch_bounds__(T_) void lr_scan_kernel(const int* __restrict__ dur,
                                                     const int* __restrict__ slen,
                                                     int* __restrict__ cum,
                                                     int* __restrict__ totals) {
  __shared__ int s[T_];
  const int b = blockIdx.x;
  const int t = threadIdx.x;
  const int L = slen[b];
  int v = (t < L) ? dur[b * T_ + t] : 0;
  s[t] = v;
  __syncthreads();
  for (int off = 1; off < T_; off <<= 1) {   // Hillis-Steele, 10 rounds
    int add = (t >= off) ? s[t - off] : 0;
    __syncthreads();
    s[t] += add;
    __syncthreads();
  }
  cum[b * T_ + t] = s[t];
  if (t == T_ - 1) totals[b] = s[t];
}

// ---------------- Kernel 2: per-position searchsorted -> tidx ----------------
// One thread per (b, pos): 65536 independent binary searches; the 10-deep
// dependent load chain (L2-resident cum, 128KB) is hidden by thread count.
// tidx = -1 marks "past total" (zero-fill).
__global__ __launch_bounds__(256) void lr_tidx_kernel(const int* __restrict__ cum,
                                                      const int* __restrict__ totals,
                                                      int* __restrict__ tidx) {
  const int gid = blockIdx.x * 256 + threadIdx.x;   // 0 .. B_*MAXLEN_-1
  const int b   = gid >> 11;                        // / MAXLEN_
  const int pos = gid & (MAXLEN_ - 1);

  int res = -1;
  if (pos < totals[b]) {
    const int* cb = cum + b * T_;
    int lo = 0, hi = T_;
    while (lo < hi) {   // first idx with cum[idx] > pos
      const int mid = (lo + hi) >> 1;
      if (cb[mid] <= pos) lo = mid + 1; else hi = mid;
    }
    res = (lo < (T_ - 1)) ? lo : (T_ - 1);
  }
  tidx[gid] = res;
}

// ---------------- Kernel 3: row gather/expand via async LDS DMA ----------------
// grid = (MAXLEN_/ROWS_PER_BLOCK, B_), 256 threads. Each block moves 4 rows of
// 4KB. All async global->LDS loads are issued first (4x MLP), one
// s_wait_asynccnt 0, then all async LDS->global stores fire with NT hint;
// S_ENDPGM's implicit wait-idle covers store completion. Bulk data never
// touches VGPRs.
__global__ __launch_bounds__(256) void lr_expand_kernel(const float4* __restrict__ x,
                                                        const int* __restrict__ tidx,
                                                        float4* __restrict__ out) {
  const int b    = blockIdx.y;
  const int pos0 = blockIdx.x * ROWS_PER_BLOCK;
  const int tid  = threadIdx.x;

  // 4 consecutive, 16B-aligned row indices in one load (block-uniform)
  const int4 t4 = *(const int4*)(tidx + b * MAXLEN_ + pos0);
  int tix[ROWS_PER_BLOCK] = {t4.x, t4.y, t4.z, t4.w};

  float4* dst0 = out + ((size_t)b * MAXLEN_ + (size_t)pos0) * C4_;
  const f32x4 zero = {0.f, 0.f, 0.f, 0.f};

#if HAS_ASYNC_LDS
  __shared__ float4 lbuf[ROWS_PER_BLOCK][256];

  // Phase 1: issue all valid row loads (async, ASYNCcnt-tracked, RT-cached)
  #pragma unroll
  for (int r = 0; r < ROWS_PER_BLOCK; ++r) {
    if (tix[r] >= 0) {
      const float4* src = x + ((size_t)b * T_ + (size_t)tix[r]) * C4_;
      __builtin_amdgcn_global_load_async_to_lds_b128(
          (b128_t*)(src + tid), (lds_b128_ptr)&lbuf[r][tid], 0, 0);
    }
  }
  WAIT_ASYNCCNT0();   // all LDS buffers filled

  // Phase 2: fire stores (async NT for gathered rows, NT b128 zeros otherwise)
  #pragma unroll
  for (int r = 0; r < ROWS_PER_BLOCK; ++r) {
    float4* dst = dst0 + (size_t)r * C4_;
    if (tix[r] >= 0) {
      __builtin_amdgcn_global_store_async_from_lds_b128(
          (b128_t*)(dst + tid), (lds_b128_ptr)&lbuf[r][tid], 0, CPOL_STORE_NT);
    } else {
      __builtin_nontemporal_store(zero, (f32x4*)(dst + tid));
    }
  }
  // no trailing wait needed: S_ENDPGM performs an implicit wait-idle
#else
  #pragma unroll
  for (int r = 0; r < ROWS_PER_BLOCK; ++r) {
    float4* dst = dst0 + (size_t)r * C4_;
    if (tix[r] >= 0) {
      const float4* src = x + ((size_t)b * T_ + (size_t)tix[r]) * C4_;
      const f32x4 v = *(const f32x4*)(src + tid);
      __builtin_nontemporal_store(v, (f32x4*)(dst + tid));
    } else {
      __builtin_nontemporal_store(zero, (f32x4*)(dst + tid));
    }
  }
#endif
}

// -------------------- launcher --------------------
extern "C" void kernel_launch(void* const* d_in, const int* in_sizes, int n_in,
                              void* d_out, int out_size, void* d_ws, size_t ws_size,
                              hipStream_t stream) {
  (void)in_sizes; (void)n_in; (void)out_size; (void)ws_size;

  const float* x    = (const float*)d_in[0];   // (B, T, C) fp32
  const int*   dur  = (const int*)d_in[1];     // (B, T) int32
  const int*   slen = (const int*)d_in[2];     // (B,) int32
  // d_in[3] = max_len scalar (compile-time constant MAXLEN_)

  // workspace layout: cum[B*T] | totals[B] | tidx[B*MAXLEN]  (~384 KB)
  int* cum    = (int*)d_ws;
  int* totals = cum + B_ * T_;
  int* tidx   = totals + 32;            // 16B-aligned offset

  lr_scan_kernel<<<B_, T_, 0, stream>>>(dur, slen, cum, totals);

  lr_tidx_kernel<<<(B_ * MAXLEN_) / 256, 256, 0, stream>>>(cum, totals, tidx);

  dim3 grid(MAXLEN_ / ROWS_PER_BLOCK, B_);
  lr_expand_kernel<<<grid, 256, 0, stream>>>((const float4*)x, tidx,
                                             (float4*)d_out);
}